// HeteroAttentionLayer_89936615178982
// MI455X (gfx1250) — compile-verified
//
#include <hip/hip_runtime.h>

#define NH   8
#define HID  128
#define EDIM 64
#define NKP  (HID + 4)    // padded K-stride for transposed W in LDS (node)
#define EKP  (EDIM + 4)   // padded K-stride for transposed W_e in LDS (edge)

typedef __attribute__((ext_vector_type(2))) float v2f;
typedef __attribute__((ext_vector_type(8))) float v8f;

__device__ __forceinline__ void atomicMaxF32(float* addr, float val) {
  // monotonic int-mapping: non-negative -> signed max, negative -> unsigned min.
  // Requires init to -inf.
  if (val >= 0.0f) {
    atomicMax((int*)addr, __float_as_int(val));
  } else {
    atomicMin((unsigned int*)addr, (unsigned int)__float_as_int(val));
  }
}

// ---------------------------------------------------------------------------
// init: out = 0, maxbuf = -inf, sumbuf = 0
// ---------------------------------------------------------------------------
__global__ void __launch_bounds__(256)
ha_init_kernel(float* __restrict__ out, float* __restrict__ maxbuf,
               float* __restrict__ sumbuf, int nOut, int nSeg) {
  int i = blockIdx.x * 256 + threadIdx.x;
  if (i < nOut) out[i] = 0.0f;
  if (i < nSeg) {
    maxbuf[i] = -__builtin_huge_valf();
    sumbuf[i] = 0.0f;
  }
}

// ---------------------------------------------------------------------------
// Node projections. Block = 128 rows x 128 cols.
// W staged TRANSPOSED in LDS: sWt[col*NKP + k] -> each B operand is one
// conflict-free ds_load_b64 straight into an aligned VGPR pair.
// Wave w owns M-tile w, sweeps 8 N-tiles with 8 independent accumulators.
// ---------------------------------------------------------------------------
__global__ void __launch_bounds__(256)
ha_node_proj_kernel(const float* __restrict__ q_feat,
                    const float* __restrict__ kv_feat,
                    const float* __restrict__ Wq, const float* __restrict__ Wk,
                    const float* __restrict__ Wv,
                    float* __restrict__ Q, float* __restrict__ K,
                    float* __restrict__ V, int Nq, int Nkv) {
  __shared__ __align__(16) float sWt[HID * NKP];  // 66 KB of 320 KB WGP LDS

  const float* X;
  const float* W;
  float* O;
  int N;
  if (blockIdx.y == 0)      { X = q_feat;  W = Wq; O = Q; N = Nq;  }
  else if (blockIdx.y == 1) { X = kv_feat; W = Wk; O = K; N = Nkv; }
  else                      { X = kv_feat; W = Wv; O = V; N = Nkv; }

  // Transposed stage: thread grabs 4 consecutive k for one col, writes float4.
  for (int idx = threadIdx.x; idx < (HID * HID) / 4; idx += 256) {
    const int c  = idx & (HID - 1);
    const int k4 = (idx >> 7) << 2;
    float4 w;
    w.x = W[(k4 + 0) * HID + c];
    w.y = W[(k4 + 1) * HID + c];
    w.z = W[(k4 + 2) * HID + c];
    w.w = W[(k4 + 3) * HID + c];
    *(float4*)&sWt[c * NKP + k4] = w;
  }
  __syncthreads();

  const int lane = threadIdx.x & 31;
  const int wv   = threadIdx.x >> 5;
  const int m0   = blockIdx.x * 128 + wv * 16;    // this wave's M-tile
  if (m0 >= N) return;

  const int hf  = lane >> 4;                      // lane half (K/M split)
  const int l16 = lane & 15;
  const int arow = min(m0 + l16, N - 1);
  const float* Arow = X + (size_t)arow * HID;

  v8f c[8] = {};
#pragma unroll 2
  for (int kk = 0; kk < HID; kk += 4) {
    const int kb = kk + hf * 2;
    const v2f a = *(const v2f*)(Arow + kb);
    const float* bp = &sWt[l16 * NKP + kb];       // col = nt*16 + l16
#pragma unroll
    for (int nt = 0; nt < 8; ++nt) {
      const v2f b = *(const v2f*)(bp + nt * 16 * NKP);
      c[nt] = __builtin_amdgcn_wmma_f32_16x16x4_f32(false, a, false, b,
                                                    (short)0, c[nt],
                                                    false, false);
    }
  }

  if (m0 + 16 <= N) {                             // branch-free fast path
#pragma unroll
    for (int r = 0; r < 8; ++r) {
      float* orow = O + (size_t)(m0 + hf * 8 + r) * HID + l16;
#pragma unroll
      for (int nt = 0; nt < 8; ++nt) orow[nt * 16] = c[nt][r];
    }
  } else {                                        // guarded tail
    for (int r = 0; r < 8; ++r) {
      const int row = m0 + hf * 8 + r;
      if (row < N) {
        float* orow = O + (size_t)row * HID + l16;
        for (int nt = 0; nt < 8; ++nt) orow[nt * 16] = c[nt][r];
      }
    }
  }
}

// ---------------------------------------------------------------------------
// Pass 1: block = 128 edges; W_e staged transposed (34 KB). Wave w owns 16
// edges, computes E_proj for all 8 heads (8 accumulators), then per-edge q.k
// dots via 16-lane shfl_xor reductions, atomicMax segment max (L2-resident).
// Gather rows prefetched (global_prefetch_b8) before the WMMA loop.
// ---------------------------------------------------------------------------
__global__ void __launch_bounds__(256)
ha_edge_logits_kernel(const float* __restrict__ edge_attr,
                      const float* __restrict__ We,
                      const int* __restrict__ src_idx,
                      const int* __restrict__ dst_idx,
                      const float* __restrict__ Q, const float* __restrict__ K,
                      float* __restrict__ logits, float* __restrict__ maxbuf,
                      int E) {
  __shared__ __align__(16) float sWt[HID * EKP];  // 34 KB
  for (int idx = threadIdx.x; idx < (EDIM * HID) / 4; idx += 256) {
    const int c  = idx & (HID - 1);
    const int k4 = (idx >> 7) << 2;
    float4 w;
    w.x = We[(k4 + 0) * HID + c];
    w.y = We[(k4 + 1) * HID + c];
    w.z = We[(k4 + 2) * HID + c];
    w.w = We[(k4 + 3) * HID + c];
    *(float4*)&sWt[c * EKP + k4] = w;
  }
  __syncthreads();

  const int lane = threadIdx.x & 31;
  const int wv   = threadIdx.x >> 5;
  const int e0   = blockIdx.x * 128 + wv * 16;    // this wave's 16 edges
  if (e0 >= E) return;

  const int hf  = lane >> 4;
  const int l16 = lane & 15;
  const int aedge = min(e0 + l16, E - 1);
  const float* Arow = edge_attr + (size_t)aedge * EDIM;

  // Warm the gather rows while WMMA runs (edge via l16, row-half via hf).
  {
    const int ps = src_idx[aedge];
    const int pd = dst_idx[aedge];
    __builtin_prefetch(&K[(size_t)ps * HID + hf * 64], 0, 3);
    __builtin_prefetch(&Q[(size_t)pd * HID + hf * 64], 0, 3);
  }

  v8f c[8] = {};
#pragma unroll 2
  for (int kk = 0; kk < EDIM; kk += 4) {
    const int kb = kk + hf * 2;
    const v2f a = *(const v2f*)(Arow + kb);
    const float* bp = &sWt[l16 * EKP + kb];
#pragma unroll
    for (int nt = 0; nt < 8; ++nt) {
      const v2f b = *(const v2f*)(bp + nt * 16 * EKP);
      c[nt] = __builtin_amdgcn_wmma_f32_16x16x4_f32(false, a, false, b,
                                                    (short)0, c[nt],
                                                    false, false);
    }
  }

  for (int r = 0; r < 8; ++r) {
    const int e = e0 + hf * 8 + r;
    if (e >= E) break;
    const int s = src_idx[e];
    const int d = dst_idx[e];
    const float* Krow = K + (size_t)s * HID + l16;
    const float* Qrow = Q + (size_t)d * HID + l16;
#pragma unroll
    for (int h = 0; h < NH; ++h) {
      float p = Qrow[h * 16] * (Krow[h * 16] + c[h][r]);
      p += __shfl_xor(p, 1, 32);
      p += __shfl_xor(p, 2, 32);
      p += __shfl_xor(p, 4, 32);
      p += __shfl_xor(p, 8, 32);
      if (l16 == 0) {
        const float logit = p * 0.25f;            // / sqrt(16)
        logits[(size_t)e * NH + h] = logit;
        atomicMaxF32(&maxbuf[(size_t)d * NH + h], logit);
      }
    }
  }
}

// ---------------------------------------------------------------------------
// Pass 2: exp_l = exp(logit - max[dst]); sumbuf[dst,h] += exp_l (in place)
// ---------------------------------------------------------------------------
__global__ void __launch_bounds__(256)
ha_edge_exp_kernel(const float* __restrict__ maxbuf,
                   const int* __restrict__ dst_idx,
                   float* __restrict__ logits, float* __restrict__ sumbuf,
                   int EH) {
  const int i = blockIdx.x * 256 + threadIdx.x;
  if (i >= EH) return;
  const int e = i >> 3;                           // / NH
  const int h = i & (NH - 1);
  const int d = dst_idx[e];
  const float m  = maxbuf[(size_t)d * NH + h];
  const float ex = __expf(logits[i] - m);
  logits[i] = ex;                                 // reuse buffer for exp
  atomicAdd(&sumbuf[(size_t)d * NH + h], ex);
}

// ---------------------------------------------------------------------------
// Pass 3: same tiling; recompute E_proj (WMMA), v_e = V[src]+E_proj,
// attn = exp_l / (sum[dst]+1e-8), scatter-add attn*v_e into out.
// ---------------------------------------------------------------------------
__global__ void __launch_bounds__(256)
ha_edge_message_kernel(const float* __restrict__ edge_attr,
                       const float* __restrict__ We,
                       const int* __restrict__ src_idx,
                       const int* __restrict__ dst_idx,
                       const float* __restrict__ V,
                       const float* __restrict__ expbuf,
                       const float* __restrict__ sumbuf,
                       float* __restrict__ out, int E) {
  __shared__ __align__(16) float sWt[HID * EKP];  // 34 KB
  for (int idx = threadIdx.x; idx < (EDIM * HID) / 4; idx += 256) {
    const int c  = idx & (HID - 1);
    const int k4 = (idx >> 7) << 2;
    float4 w;
    w.x = We[(k4 + 0) * HID + c];
    w.y = We[(k4 + 1) * HID + c];
    w.z = We[(k4 + 2) * HID + c];
    w.w = We[(k4 + 3) * HID + c];
    *(float4*)&sWt[c * EKP + k4] = w;
  }
  __syncthreads();

  const int lane = threadIdx.x & 31;
  const int wv   = threadIdx.x >> 5;
  const int e0   = blockIdx.x * 128 + wv * 16;
  if (e0 >= E) return;

  const int hf  = lane >> 4;
  const int l16 = lane & 15;
  const int aedge = min(e0 + l16, E - 1);
  const float* Arow = edge_attr + (size_t)aedge * EDIM;

  {
    const int ps = src_idx[aedge];
    __builtin_prefetch(&V[(size_t)ps * HID + hf * 64], 0, 3);
    __builtin_prefetch(&expbuf[(size_t)aedge * NH], 0, 3);
  }

  v8f c[8] = {};
#pragma unroll 2
  for (int kk = 0; kk < EDIM; kk += 4) {
    const int kb = kk + hf * 2;
    const v2f a = *(const v2f*)(Arow + kb);
    const float* bp = &sWt[l16 * EKP + kb];
#pragma unroll
    for (int nt = 0; nt < 8; ++nt) {
      const v2f b = *(const v2f*)(bp + nt * 16 * EKP);
      c[nt] = __builtin_amdgcn_wmma_f32_16x16x4_f32(false, a, false, b,
                                                    (short)0, c[nt],
                                                    false, false);
    }
  }

  for (int r = 0; r < 8; ++r) {
    const int e = e0 + hf * 8 + r;
    if (e >= E) break;
    const int s = src_idx[e];
    const int d = dst_idx[e];
    const float* Vrow = V + (size_t)s * HID + l16;
    const float* exr  = expbuf + (size_t)e * NH;
    const float* smr  = sumbuf + (size_t)d * NH;
    float* orow       = out + (size_t)d * HID + l16;
#pragma unroll
    for (int h = 0; h < NH; ++h) {
      const float attn = exr[h] / (smr[h] + 1e-8f);
      const float vv   = Vrow[h * 16] + c[h][r];
      atomicAdd(&orow[h * 16], attn * vv);        // L2-resident scatter
    }
  }
}

// ---------------------------------------------------------------------------
extern "C" void kernel_launch(void* const* d_in, const int* in_sizes, int n_in,
                              void* d_out, int out_size, void* d_ws,
                              size_t ws_size, hipStream_t stream) {
  const float* q_feat    = (const float*)d_in[0];
  const float* kv_feat   = (const float*)d_in[1];
  const float* edge_attr = (const float*)d_in[2];
  const int*   src_idx   = (const int*)d_in[3];
  const int*   dst_idx   = (const int*)d_in[4];
  // d_in[5] = num_dst (device scalar) — equals Nq in this workload
  const float* W_q = (const float*)d_in[6];
  const float* W_k = (const float*)d_in[7];
  const float* W_e = (const float*)d_in[8];
  const float* W_v = (const float*)d_in[9];
  float* out = (float*)d_out;

  const int Nq      = in_sizes[0] / HID;
  const int Nkv     = in_sizes[1] / HID;
  const int E       = in_sizes[3];
  const int num_dst = Nq;

  // Workspace (floats): Q | K | V | logits/exp | maxbuf | sumbuf  (~85 MB)
  float* Q      = (float*)d_ws;
  float* K      = Q + (size_t)Nq * HID;
  float* V      = K + (size_t)Nkv * HID;
  float* logits = V + (size_t)Nkv * HID;
  float* maxbuf = logits + (size_t)E * NH;
  float* sumbuf = maxbuf + (size_t)num_dst * NH;

  const int nOut  = num_dst * HID;
  const int nSeg  = num_dst * NH;
  const int nInit = nOut > nSeg ? nOut : nSeg;
  ha_init_kernel<<<(nInit + 255) / 256, 256, 0, stream>>>(out, maxbuf, sumbuf,
                                                          nOut, nSeg);

  const int nmax = Nq > Nkv ? Nq : Nkv;
  dim3 gn((nmax + 127) / 128, 3);
  ha_node_proj_kernel<<<gn, 256, 0, stream>>>(q_feat, kv_feat, W_q, W_k, W_v,
                                              Q, K, V, Nq, Nkv);

  const int eBlocks = (E + 127) / 128;
  ha_edge_logits_kernel<<<eBlocks, 256, 0, stream>>>(
      edge_attr, W_e, src_idx, dst_idx, Q, K, logits, maxbuf, E);

  const int EH = E * NH;
  ha_edge_exp_kernel<<<(EH + 255) / 256, 256, 0, stream>>>(maxbuf, dst_idx,
                                                           logits, sumbuf, EH);

  ha_edge_message_kernel<<<eBlocks, 256, 0, stream>>>(
      edge_attr, W_e, src_idx, dst_idx, V, logits, sumbuf, out, E);
}